// DensityAwareChamferDistance_22376779612396
// MI455X (gfx1250) — compile-verified
//
#include <hip/hip_runtime.h>
#include <hip/hip_bf16.h>

typedef __attribute__((ext_vector_type(2))) float v2f;
typedef __attribute__((ext_vector_type(8))) float v8f;

#define BATCH 4
#define NPTS  8192            // N == M == 8192, power of two (>>13)
#define ROWS_PER_WAVE  16
#define WAVES_PER_BLK  8
#define ROWS_PER_BLK   (ROWS_PER_WAVE * WAVES_PER_BLK)   // 128

// ---------------------------------------------------------------------------
// NN kernel: for each point in `apts`, find min_n (|b_n|^2 - 2 a.b_n) and its
// argmin over all points in `bpts` (same batch). |a|^2 is added downstream.
// One wave owns 16 rows; A operand (pre-scaled by -2) is loop-invariant.
// D = A x B + C with C = |b_n|^2 broadcast  =>  D[m][n] = yy_n - 2 x_m.y_n.
// ---------------------------------------------------------------------------
__global__ __launch_bounds__(256)
void dacd_nn_wmma(const float* __restrict__ apts, const float* __restrict__ bpts,
                  float* __restrict__ smin_out, int* __restrict__ idx_out) {
  const int  lane = threadIdx.x & 31;
  const int  wave = threadIdx.x >> 5;
  const int  lm   = lane & 15;
  const bool hi   = lane >= 16;
  const int  b    = blockIdx.y;
  const int  row_base = blockIdx.x * ROWS_PER_BLK + wave * ROWS_PER_WAVE;

  // A tile: 16 x-points, K = {x,y,z,0}, pre-scaled by -2.
  // Layout: lanes 0-15 hold K0/K1 (VGPR0/1), lanes 16-31 hold K2/K3.
  const float* xp = apts + ((size_t)b * NPTS + row_base + lm) * 3;
  const float ax = xp[0], ay = xp[1], az = xp[2];
  v2f A;
  A.x = hi ? (-2.0f * az) : (-2.0f * ax);
  A.y = hi ? 0.0f         : (-2.0f * ay);

  float minv[8];
  int   mini[8];
#pragma unroll
  for (int r = 0; r < 8; ++r) { minv[r] = 3.402823466e38f; mini[r] = 0; }

  const float* ybase = bpts + (size_t)b * NPTS * 3;
  for (int t = 0; t < NPTS / 16; ++t) {
    const float* yp = ybase + (t * 16 + lm) * 3;
    const float y0 = yp[0], y1 = yp[1], y2 = yp[2];
    __builtin_prefetch(yp + 3 * 64, 0, 1);           // global_prefetch_b8, 4 tiles ahead
    const float yy = y0 * y0 + y1 * y1 + y2 * y2;

    v2f Bv;                                          // B tile: 4x16, same K split as A
    Bv.x = hi ? y2 : y0;
    Bv.y = hi ? 0.0f : y1;

    v8f c = {yy, yy, yy, yy, yy, yy, yy, yy};        // C[m][n] = |y_n|^2 (N = lane%16)
    // D[m][n] = yy_n - 2 * dot(x_m, y_n)   (single matrix-pipe op, K=4, fp32)
    v8f d = __builtin_amdgcn_wmma_f32_16x16x4_f32(
        /*neg_a=*/false, A, /*neg_b=*/false, Bv,
        /*c_mod=*/(short)0, c, /*reuse_a=*/false, /*reuse_b=*/false);

    const int nidx = t * 16 + lm;
#pragma unroll
    for (int r = 0; r < 8; ++r) {                    // running min/argmin per row slot
      const float s = d[r];
      if (s < minv[r]) { minv[r] = s; mini[r] = nidx; }
    }
  }

  // Cross-lane argmin within each 16-lane half (xor 8,4,2,1 stays in the half).
#pragma unroll
  for (int off = 8; off >= 1; off >>= 1) {
#pragma unroll
    for (int r = 0; r < 8; ++r) {
      const float ov = __shfl_xor(minv[r], off, 32);
      const int   oi = __shfl_xor(mini[r], off, 32);
      if (ov < minv[r] || (ov == minv[r] && oi < mini[r])) { minv[r] = ov; mini[r] = oi; }
    }
  }

  // lane 0 holds rows 0..7, lane 16 holds rows 8..15
  if (lm == 0) {
    const size_t base = (size_t)b * NPTS + row_base + (hi ? 8 : 0);
#pragma unroll
    for (int r = 0; r < 8; ++r) { smin_out[base + r] = minv[r]; idx_out[base + r] = mini[r]; }
  }
}

// ---------------------------------------------------------------------------
__global__ __launch_bounds__(256)
void dacd_zero_i32(int* __restrict__ p, int n) {
  const int i = blockIdx.x * blockDim.x + threadIdx.x;
  if (i < n) p[i] = 0;
}

// Density counts: integer scatter-add (order-independent -> deterministic).
__global__ __launch_bounds__(256)
void dacd_count(const int* __restrict__ idx, int* __restrict__ cnt) {
  const int i = blockIdx.x * blockDim.x + threadIdx.x;
  if (i < BATCH * NPTS) {
    const int b = i >> 13;
    atomicAdd(&cnt[((size_t)b << 13) + idx[i]], 1);
  }
}

// Per-point loss terms + deterministic in-block tree reduction.
__global__ __launch_bounds__(256)
void dacd_loss(const float* __restrict__ pts, const float* __restrict__ smin,
               const int* __restrict__ idx, const int* __restrict__ cnt,
               float* __restrict__ partial) {
  __shared__ float red[256];
  const int i = blockIdx.x * 256 + threadIdx.x;     // grid exactly covers B*N
  const float* p = pts + (size_t)i * 3;
  const float xx   = p[0] * p[0] + p[1] * p[1] + p[2] * p[2];
  const float dist = smin[i] + xx;                  // full squared distance
  const int   b    = i >> 13;
  const float c    = (float)cnt[((size_t)b << 13) + idx[i]];
  const float w    = 1.0f / (c + 1e-6f);            // N_LAMBDA = 1
  const float term = 1.0f - __expf(-dist * 1000.0f) * w;   // ALPHA = 1000

  red[threadIdx.x] = term;
  __syncthreads();
  for (int s = 128; s > 0; s >>= 1) {
    if (threadIdx.x < s) red[threadIdx.x] += red[threadIdx.x + s];
    __syncthreads();
  }
  if (threadIdx.x == 0) partial[blockIdx.x] = red[0];
}

// Single-block fixed-order final reduction.
__global__ __launch_bounds__(256)
void dacd_final(const float* __restrict__ p1, const float* __restrict__ p2, int n,
                float* __restrict__ out) {
  __shared__ float red[256];
  float s = 0.0f;
  for (int i = threadIdx.x; i < n; i += 256) s += p1[i] + p2[i];
  red[threadIdx.x] = s;
  __syncthreads();
  for (int k = 128; k > 0; k >>= 1) {
    if (threadIdx.x < k) red[threadIdx.x] += red[threadIdx.x + k];
    __syncthreads();
  }
  if (threadIdx.x == 0)
    out[0] = red[0] * (1.0f / (2.0f * BATCH * NPTS));  // mean((l1+l2)/2) over batch
}

// ---------------------------------------------------------------------------
extern "C" void kernel_launch(void* const* d_in, const int* in_sizes, int n_in,
                              void* d_out, int out_size, void* d_ws, size_t ws_size,
                              hipStream_t stream) {
  const float* x = (const float*)d_in[0];   // [4, 8192, 3] f32
  const float* y = (const float*)d_in[1];   // [4, 8192, 3] f32

  const size_t BN = (size_t)BATCH * NPTS;   // 32768
  char* ws = (char*)d_ws;
  float* s1    = (float*)ws; ws += BN * 4;  // min_n (yy - 2xy) for x->y
  int*   i1    = (int*)  ws; ws += BN * 4;  // argmin
  float* s2    = (float*)ws; ws += BN * 4;  // y->x
  int*   i2    = (int*)  ws; ws += BN * 4;
  int*   c1    = (int*)  ws; ws += BN * 4;  // density counts (c1,c2 contiguous)
  int*   c2    = (int*)  ws; ws += BN * 4;
  float* part1 = (float*)ws; ws += 128 * 4;
  float* part2 = (float*)ws;

  // 1) zero both count arrays (contiguous) — d_ws is poisoned, must init
  dacd_zero_i32<<<(2 * (int)BN + 255) / 256, 256, 0, stream>>>(c1, 2 * (int)BN);

  // 2) nearest-neighbor passes (WMMA) — dist matrix is symmetric-transpose,
  //    so the y->x pass is the same kernel with operands swapped.
  dim3 grid(NPTS / ROWS_PER_BLK, BATCH);    // (64, 4)
  dacd_nn_wmma<<<grid, 256, 0, stream>>>(x, y, s1, i1);
  dacd_nn_wmma<<<grid, 256, 0, stream>>>(y, x, s2, i2);

  // 3) density counts
  dacd_count<<<(int)(BN + 255) / 256, 256, 0, stream>>>(i1, c1);
  dacd_count<<<(int)(BN + 255) / 256, 256, 0, stream>>>(i2, c2);

  // 4) loss terms + block partials (128 blocks each, exact cover)
  dacd_loss<<<(int)(BN / 256), 256, 0, stream>>>(x, s1, i1, c1, part1);
  dacd_loss<<<(int)(BN / 256), 256, 0, stream>>>(y, s2, i2, c2, part2);

  // 5) final scalar
  dacd_final<<<1, 256, 0, stream>>>(part1, part2, 128, (float*)d_out);
}